// GATEncoder_2241972928922
// MI455X (gfx1250) — compile-verified
//
#include <hip/hip_runtime.h>
#include <hip/hip_bf16.h>
#include <math.h>

// ---------------- problem constants ----------------
#define NN   50000
#define NE   150000
#define NE2  200000   // NE + NN (self loops)
#define DIN  48
#define DH   64
#define DE   16
#define HEADS 10
#define HF   640      // HEADS * DH
#define NLAYERS 3
#define DOUT 128
#define NG   4096

typedef __attribute__((ext_vector_type(2))) float v2f;
typedef __attribute__((ext_vector_type(8))) float v8f;

// ---------------- WMMA helper (fp32, 16x16x4) ----------------
__device__ __forceinline__ v8f wmma_f32_16x16x4(v2f a, v2f b, v8f c) {
    return __builtin_amdgcn_wmma_f32_16x16x4_f32(false, a, false, b, (short)0, c, false, false);
}

// Register-blocked fp32 GEMM: C[M x Nc] = A[M x K] @ B[K x Nc] (+ bias[Nc]).
// One block per 16-row m-tile. Each wave owns TPW consecutive 16-col n-tiles;
// the A fragment is loaded once per K-step and reused across TPW WMMAs, so A
// is read exactly once per strip (B stays hot in WGP$/L2).
// Requires: blockDim.x/32 * TPW == Nc/16.
template <int TPW>
__global__ __launch_bounds__(256) void gemm_wmma_kernel(
    const float* __restrict__ A, const float* __restrict__ B,
    const float* __restrict__ bias, float* __restrict__ C,
    int K, int Nc)
{
    const int lane  = threadIdx.x & 31;
    const int wave  = threadIdx.x >> 5;
    const int mTile = blockIdx.x;
    const int l16   = lane & 15;
    const int half  = lane >> 4;
    const int kb    = half * 2;                 // K sub-offset for this half-wave
    const int row   = mTile * 16 + l16;
    const int col0  = wave * TPW * 16 + l16;    // first owned column

    v8f acc[TPW];
#pragma unroll
    for (int t = 0; t < TPW; ++t) acc[t] = (v8f){};

    const float* Arow = A + (long long)row * K;
    for (int k = 0; k < K; k += 4) {
        v2f a;
        a.x = Arow[k + kb];
        a.y = Arow[k + kb + 1];
        const float* B0 = B + (long long)(k + kb)     * Nc + col0;
        const float* B1 = B + (long long)(k + kb + 1) * Nc + col0;
#pragma unroll
        for (int t = 0; t < TPW; ++t) {
            v2f b;
            b.x = B0[t * 16];
            b.y = B1[t * 16];
            acc[t] = wmma_f32_16x16x4(a, b, acc[t]);
        }
    }

#pragma unroll
    for (int t = 0; t < TPW; ++t) {
        const int col = col0 + t * 16;
        const float bv = bias ? bias[col] : 0.0f;
#pragma unroll
        for (int v = 0; v < 8; ++v) {
            int r = mTile * 16 + v + half * 8;
            C[(long long)r * Nc + col] = acc[t][v] + bv;
        }
    }
}

// ---------------- utility kernels ----------------
__global__ void fill_kernel(float* __restrict__ p, float v, long long n) {
    long long i = (long long)blockIdx.x * blockDim.x + threadIdx.x;
    if (i < n) p[i] = v;
}

// accumulate incoming-edge count and edge_attr sum per destination node
__global__ void edge_accum_kernel(const float* __restrict__ edge_attr,
                                  const int* __restrict__ edge_index,
                                  float* __restrict__ cnt, float* __restrict__ loop_attr) {
    int e = blockIdx.x * blockDim.x + threadIdx.x;
    if (e >= NE) return;
    int d = edge_index[NE + e];
    atomicAdd(&cnt[d], 1.0f);
#pragma unroll
    for (int j = 0; j < DE; ++j)
        atomicAdd(&loop_attr[(long long)d * DE + j], edge_attr[(long long)e * DE + j]);
}

__global__ void loop_div_kernel(float* __restrict__ loop_attr, const float* __restrict__ cnt) {
    long long i = (long long)blockIdx.x * blockDim.x + threadIdx.x;
    if (i >= (long long)NN * DE) return;
    int n = (int)(i / DE);
    loop_attr[i] /= fmaxf(cnt[n], 1.0f);
}

// pre-contract attention vectors into the weight matrices:
// w_s[d][h] = sum_f Ws[d, h*64+f]*att_src[h][f]   (d<64)   (same for w_d)
// w_e[j][h] = sum_f We[j, h*64+f]*att_edge[h][f]  (j<16)
__global__ void contract_kernel(const float* __restrict__ Wsi,
                                const float* __restrict__ asrc,
                                const float* __restrict__ adst,
                                const float* __restrict__ Wei,
                                const float* __restrict__ aedge,
                                float* __restrict__ w_s, float* __restrict__ w_d,
                                float* __restrict__ w_e) {
    int t = threadIdx.x;
    if (t < DH * HEADS) {
        int d = t / HEADS, h = t % HEADS;
        float ss = 0.f, sd = 0.f;
        for (int f = 0; f < DH; ++f) {
            float wv = Wsi[(long long)d * HF + h * DH + f];
            ss += wv * asrc[h * DH + f];
            sd += wv * adst[h * DH + f];
        }
        w_s[d * HEADS + h] = ss;
        w_d[d * HEADS + h] = sd;
    } else if (t < DH * HEADS + DE * HEADS) {
        int u = t - DH * HEADS;
        int j = u / HEADS, h = u % HEADS;
        float se = 0.f;
        for (int f = 0; f < DH; ++f)
            se += Wei[(long long)j * HF + h * DH + f] * aedge[h * DH + f];
        w_e[j * HEADS + h] = se;
    }
}

// a_s[n,h] = h[n,:] . w_s[:,h]   ;   a_d likewise
__global__ void asd_kernel(const float* __restrict__ hbuf,
                           const float* __restrict__ w_s, const float* __restrict__ w_d,
                           float* __restrict__ a_s, float* __restrict__ a_d) {
    long long i = (long long)blockIdx.x * blockDim.x + threadIdx.x;
    if (i >= (long long)NN * HEADS) return;
    int n = (int)(i / HEADS), h = (int)(i % HEADS);
    float ss = 0.f, sd = 0.f;
    const float* hr = hbuf + (long long)n * DH;
#pragma unroll 8
    for (int d = 0; d < DH; ++d) {
        float hv = hr[d];
        ss += hv * w_s[d * HEADS + h];
        sd += hv * w_d[d * HEADS + h];
    }
    a_s[i] = ss;
    a_d[i] = sd;
}

// a_e[e2,h] = ea2[e2,:] . w_e[:,h]
__global__ void ae_kernel(const float* __restrict__ edge_attr,
                          const float* __restrict__ loop_attr,
                          const float* __restrict__ w_e, float* __restrict__ a_e) {
    long long i = (long long)blockIdx.x * blockDim.x + threadIdx.x;
    if (i >= (long long)NE2 * HEADS) return;
    int e2 = (int)(i / HEADS), h = (int)(i % HEADS);
    const float* attr = (e2 < NE) ? (edge_attr + (long long)e2 * DE)
                                  : (loop_attr + (long long)(e2 - NE) * DE);
    float s = 0.f;
#pragma unroll
    for (int j = 0; j < DE; ++j) s += attr[j] * w_e[j * HEADS + h];
    a_e[i] = s;
}

__device__ __forceinline__ void atomicMaxFloat(float* addr, float v) {
    if (v >= 0.0f) atomicMax((int*)addr, __float_as_int(v));
    else           atomicMin((unsigned int*)addr, __float_as_uint(v));
}

// alpha = leaky_relu(a_s[src]+a_d[dst]+a_e, 0.2); segment max into m[dst,h]
__global__ void alpha_kernel(const float* __restrict__ a_s, const float* __restrict__ a_d,
                             const float* __restrict__ a_e, const int* __restrict__ edge_index,
                             float* __restrict__ alpha, float* __restrict__ m) {
    long long i = (long long)blockIdx.x * blockDim.x + threadIdx.x;
    if (i >= (long long)NE2 * HEADS) return;
    int e2 = (int)(i / HEADS), h = (int)(i % HEADS);
    int s, d;
    if (e2 < NE) { s = edge_index[e2]; d = edge_index[NE + e2]; }
    else         { s = d = e2 - NE; }
    float a = a_s[(long long)s * HEADS + h] + a_d[(long long)d * HEADS + h] + a_e[i];
    a = (a > 0.f) ? a : 0.2f * a;
    alpha[i] = a;
    atomicMaxFloat(&m[(long long)d * HEADS + h], a);
}

// e = exp(alpha - m[dst]); accumulate z[dst]
__global__ void expsum_kernel(float* __restrict__ alpha, const float* __restrict__ m,
                              const int* __restrict__ edge_index, float* __restrict__ z) {
    long long i = (long long)blockIdx.x * blockDim.x + threadIdx.x;
    if (i >= (long long)NE2 * HEADS) return;
    int e2 = (int)(i / HEADS), h = (int)(i % HEADS);
    int d = (e2 < NE) ? edge_index[NE + e2] : (e2 - NE);
    float ev = __expf(alpha[i] - m[(long long)d * HEADS + h]);
    alpha[i] = ev;
    atomicAdd(&z[(long long)d * HEADS + h], ev);
}

// att = e / z[dst]
__global__ void attdiv_kernel(float* __restrict__ alpha, const float* __restrict__ z,
                              const int* __restrict__ edge_index) {
    long long i = (long long)blockIdx.x * blockDim.x + threadIdx.x;
    if (i >= (long long)NE2 * HEADS) return;
    int e2 = (int)(i / HEADS), h = (int)(i % HEADS);
    int d = (e2 < NE) ? edge_index[NE + e2] : (e2 - NE);
    alpha[i] = alpha[i] / z[(long long)d * HEADS + h];
}

// agg[dst,h,f] += xp[src,h,f] * att[e2,h]  ; thread per (e2, f)
__global__ void agg_kernel(const float* __restrict__ alpha, const float* __restrict__ xp,
                           const int* __restrict__ edge_index, float* __restrict__ agg) {
    long long i = (long long)blockIdx.x * blockDim.x + threadIdx.x;
    if (i >= (long long)NE2 * DH) return;
    int e2 = (int)(i / DH), f = (int)(i % DH);
    int s, d;
    if (e2 < NE) { s = edge_index[e2]; d = edge_index[NE + e2]; }
    else         { s = d = e2 - NE; }
    const float* xpr = xp + (long long)s * HF + f;
    float* aggr = agg + (long long)d * HF + f;
    const float* att = alpha + (long long)e2 * HEADS;
#pragma unroll
    for (int h = 0; h < HEADS; ++h)
        atomicAdd(&aggr[h * DH], xpr[h * DH] * att[h]);
}

// head-mean + bias + batchnorm + exact GELU + residual into h
__global__ void post_kernel(const float* __restrict__ agg, float* __restrict__ hbuf,
                            const float* __restrict__ bias, const float* __restrict__ gamma,
                            const float* __restrict__ beta, const float* __restrict__ mean,
                            const float* __restrict__ var) {
    long long i = (long long)blockIdx.x * blockDim.x + threadIdx.x;
    if (i >= (long long)NN * DH) return;
    int n = (int)(i / DH), d = (int)(i % DH);
    float acc = 0.f;
    const float* ar = agg + (long long)n * HF + d;
#pragma unroll
    for (int h = 0; h < HEADS; ++h) acc += ar[h * DH];
    float xc = acc * (1.0f / HEADS) + bias[d];
    xc = (xc - mean[d]) * gamma[d] * rsqrtf(var[d] + 1e-5f) + beta[d];
    xc = 0.5f * xc * (1.0f + erff(xc * 0.70710678118654752f));   // exact GELU
    hbuf[i] += xc;
}

// graph mean-pool: accumulate sums into d_out and counts into gcnt
__global__ void pool_kernel(const float* __restrict__ outbuf, const int* __restrict__ batch_ids,
                            float* __restrict__ gsum, float* __restrict__ gcnt) {
    long long i = (long long)blockIdx.x * blockDim.x + threadIdx.x;
    if (i >= (long long)NN * DOUT) return;
    int n = (int)(i / DOUT), c = (int)(i % DOUT);
    int b = batch_ids[n];
    atomicAdd(&gsum[(long long)b * DOUT + c], outbuf[i]);
    if (c == 0) atomicAdd(&gcnt[b], 1.0f);
}

__global__ void pooldiv_kernel(float* __restrict__ gsum, const float* __restrict__ gcnt) {
    long long i = (long long)blockIdx.x * blockDim.x + threadIdx.x;
    if (i >= (long long)NG * DOUT) return;
    int g = (int)(i / DOUT);
    gsum[i] /= fmaxf(gcnt[g], 1.0f);
}

// ---------------- workspace layout (in floats) ----------------
#define OFF_CNT   0LL
#define OFF_LOOP  (OFF_CNT  + NN)
#define OFF_H     (OFF_LOOP + (long long)NN * DE)
#define OFF_XP    (OFF_H    + (long long)NN * DH)
#define OFF_AGG   (OFF_XP   + (long long)NN * HF)
#define OFF_AS    (OFF_AGG  + (long long)NN * HF)
#define OFF_AD    (OFF_AS   + (long long)NN * HEADS)
#define OFF_AE    (OFF_AD   + (long long)NN * HEADS)
#define OFF_ALPHA (OFF_AE   + (long long)NE2 * HEADS)
#define OFF_M     (OFF_ALPHA+ (long long)NE2 * HEADS)
#define OFF_Z     (OFF_M    + (long long)NN * HEADS)
#define OFF_WS    (OFF_Z    + (long long)NN * HEADS)
#define OFF_WD    (OFF_WS   + (long long)DH * HEADS)
#define OFF_WE    (OFF_WD   + (long long)DH * HEADS)
#define OFF_OUT   (OFF_WE   + (long long)DE * HEADS)
#define OFF_GCNT  (OFF_OUT  + (long long)NN * DOUT)

static inline int blocks1d(long long n, int bs) { return (int)((n + bs - 1) / bs); }

extern "C" void kernel_launch(void* const* d_in, const int* in_sizes, int n_in,
                              void* d_out, int out_size, void* d_ws, size_t ws_size,
                              hipStream_t stream) {
    const float* x         = (const float*)d_in[0];
    const float* edge_attr = (const float*)d_in[1];
    const float* W_emb     = (const float*)d_in[2];
    const float* b_emb     = (const float*)d_in[3];
    const float* Ws        = (const float*)d_in[4];
    const float* att_src   = (const float*)d_in[5];
    const float* att_dst   = (const float*)d_in[6];
    const float* att_edge  = (const float*)d_in[7];
    const float* W_edge    = (const float*)d_in[8];
    const float* bias      = (const float*)d_in[9];
    const float* bn_gamma  = (const float*)d_in[10];
    const float* bn_beta   = (const float*)d_in[11];
    const float* bn_mean   = (const float*)d_in[12];
    const float* bn_var    = (const float*)d_in[13];
    const float* W_out     = (const float*)d_in[14];
    const float* b_out     = (const float*)d_in[15];
    const int*   edge_index= (const int*)d_in[16];
    const int*   batch_ids = (const int*)d_in[17];

    float* ws   = (float*)d_ws;
    float* gsum = (float*)d_out;

    float* cnt  = ws + OFF_CNT;
    float* loop = ws + OFF_LOOP;
    float* hbuf = ws + OFF_H;
    float* xp   = ws + OFF_XP;
    float* agg  = ws + OFF_AGG;
    float* a_s  = ws + OFF_AS;
    float* a_d  = ws + OFF_AD;
    float* a_e  = ws + OFF_AE;
    float* alpha= ws + OFF_ALPHA;
    float* m    = ws + OFF_M;
    float* z    = ws + OFF_Z;
    float* w_s  = ws + OFF_WS;
    float* w_d  = ws + OFF_WD;
    float* w_e  = ws + OFF_WE;
    float* outb = ws + OFF_OUT;
    float* gcnt = ws + OFF_GCNT;

    const int BS = 256;

    // ---- self-loop edge attr (scatter mean of incoming edge_attr) ----
    fill_kernel<<<blocks1d(NN, BS), BS, 0, stream>>>(cnt, 0.f, NN);
    fill_kernel<<<blocks1d((long long)NN * DE, BS), BS, 0, stream>>>(loop, 0.f, (long long)NN * DE);
    edge_accum_kernel<<<blocks1d(NE, BS), BS, 0, stream>>>(edge_attr, edge_index, cnt, loop);
    loop_div_kernel<<<blocks1d((long long)NN * DE, BS), BS, 0, stream>>>(loop, cnt);

    // ---- embedding: h = x @ W_emb + b_emb  (M=50000, K=48, Nc=64) ----
    // 4 waves x TPW=1 covers 4 n-tiles
    gemm_wmma_kernel<1><<<NN / 16, 128, 0, stream>>>(x, W_emb, b_emb, hbuf, DIN, DH);

    for (int i = 0; i < NLAYERS; ++i) {
        const float* Wsi  = Ws       + (long long)i * DH * HF;
        const float* asrc = att_src  + (long long)i * HEADS * DH;
        const float* adst = att_dst  + (long long)i * HEADS * DH;
        const float* aedg = att_edge + (long long)i * HEADS * DH;
        const float* Wei  = W_edge   + (long long)i * DE * HF;
        const float* bi   = bias     + (long long)i * DH;
        const float* gi   = bn_gamma + (long long)i * DH;
        const float* bti  = bn_beta  + (long long)i * DH;
        const float* mi   = bn_mean  + (long long)i * DH;
        const float* vi   = bn_var   + (long long)i * DH;

        // contract attention vectors into small matrices
        contract_kernel<<<1, 1024, 0, stream>>>(Wsi, asrc, adst, Wei, aedg, w_s, w_d, w_e);

        // xp = h @ Ws[i]   (M=50000, K=64, Nc=640): 8 waves x TPW=5 -> 40 n-tiles,
        // A strip read once, 5 WMMAs per A fragment
        gemm_wmma_kernel<5><<<NN / 16, 256, 0, stream>>>(hbuf, Wsi, nullptr, xp, DH, HF);

        // per-node / per-edge attention logits
        asd_kernel<<<blocks1d((long long)NN * HEADS, BS), BS, 0, stream>>>(hbuf, w_s, w_d, a_s, a_d);
        ae_kernel<<<blocks1d((long long)NE2 * HEADS, BS), BS, 0, stream>>>(edge_attr, loop, w_e, a_e);

        // segment softmax
        fill_kernel<<<blocks1d((long long)NN * HEADS, BS), BS, 0, stream>>>(m, -1e30f, (long long)NN * HEADS);
        fill_kernel<<<blocks1d((long long)NN * HEADS, BS), BS, 0, stream>>>(z, 0.f, (long long)NN * HEADS);
        fill_kernel<<<blocks1d((long long)NN * HF, BS), BS, 0, stream>>>(agg, 0.f, (long long)NN * HF);

        alpha_kernel<<<blocks1d((long long)NE2 * HEADS, BS), BS, 0, stream>>>(a_s, a_d, a_e, edge_index, alpha, m);
        expsum_kernel<<<blocks1d((long long)NE2 * HEADS, BS), BS, 0, stream>>>(alpha, m, edge_index, z);
        attdiv_kernel<<<blocks1d((long long)NE2 * HEADS, BS), BS, 0, stream>>>(alpha, z, edge_index);

        // weighted scatter aggregation
        agg_kernel<<<blocks1d((long long)NE2 * DH, BS), BS, 0, stream>>>(alpha, xp, edge_index, agg);

        // head mean + bias + BN + GELU + residual
        post_kernel<<<blocks1d((long long)NN * DH, BS), BS, 0, stream>>>(agg, hbuf, bi, gi, bti, mi, vi);
    }

    // ---- readout: out = h @ W_out + b_out  (M=50000, K=64, Nc=128): 8 waves x TPW=1 ----
    gemm_wmma_kernel<1><<<NN / 16, 256, 0, stream>>>(hbuf, W_out, b_out, outb, DH, DOUT);

    // ---- global mean pool into d_out ----
    fill_kernel<<<blocks1d((long long)NG * DOUT, BS), BS, 0, stream>>>(gsum, 0.f, (long long)NG * DOUT);
    fill_kernel<<<blocks1d(NG, BS), BS, 0, stream>>>(gcnt, 0.f, NG);
    pool_kernel<<<blocks1d((long long)NN * DOUT, BS), BS, 0, stream>>>(outb, batch_ids, gsum, gcnt);
    pooldiv_kernel<<<blocks1d((long long)NG * DOUT, BS), BS, 0, stream>>>(gsum, gcnt);
}